// GraphPool_31147102830924
// MI455X (gfx1250) — compile-verified
//
#include <hip/hip_runtime.h>

// Problem constants (match reference)
#define NPER  4096      // points per cloud
#define NB    16        // clouds
#define MSAMP 1024      // samples per cloud (ratio 0.25)
#define DFEAT 512       // feature dim

typedef __attribute__((address_space(3))) float lds_f32;
typedef float v4f __attribute__((ext_vector_type(4)));   // native vector for NT ops

// ---------------------------------------------------------------------------
// Kernel 1: farthest point sampling, one workgroup per cloud (wave32, 8 waves)
// ---------------------------------------------------------------------------
__global__ __launch_bounds__(256)
void fps_kernel(const float* __restrict__ pos, int* __restrict__ idx_out) {
    __shared__ float spos[NPER * 3];                // 48 KB: whole cloud in LDS
    __shared__ unsigned long long skeys[2][8];      // double-buffered wave keys

    const int b   = blockIdx.x;
    const int tid = threadIdx.x;

    // ---- CDNA5 async copy: global -> LDS (no VGPR staging) ----------------
    // 12288 floats = 3072 x 16B chunks; 256 lanes x 12 iterations of B128.
    const float* gbase = pos + (size_t)b * (NPER * 3);
    unsigned lbase = (unsigned)(unsigned long long)(lds_f32*)&spos[0];
#pragma unroll
    for (int i = 0; i < 12; ++i) {
        int e = (tid + i * 256) * 4;                // float index of 16B chunk
        unsigned laddr = lbase + (unsigned)e * 4u;
        unsigned long long gaddr = (unsigned long long)(gbase + e);
        asm volatile("global_load_async_to_lds_b128 %0, %1, off"
                     :: "v"(laddr), "v"(gaddr) : "memory");
    }
    asm volatile("s_wait_asynccnt 0x0" ::: "memory");
    __syncthreads();

    // ---- own 16 points in registers ---------------------------------------
    float px[16], py[16], pz[16], dist[16];
#pragma unroll
    for (int i = 0; i < 16; ++i) {
        int p = tid + i * 256;
        px[i] = spos[p * 3 + 0];
        py[i] = spos[p * 3 + 1];
        pz[i] = spos[p * 3 + 2];
        dist[i] = 1e30f;
    }

    int last = 0;
    if (tid == 0) idx_out[b * MSAMP] = b * NPER;    // deterministic start at 0

    for (int m = 1; m < MSAMP; ++m) {
        // broadcast read of pos[last] from LDS (same address, all lanes)
        float lx = spos[last * 3 + 0];
        float ly = spos[last * 3 + 1];
        float lz = spos[last * 3 + 2];

        float    bestd = -1.0f;
        unsigned bestp = 0u;
#pragma unroll
        for (int i = 0; i < 16; ++i) {
            float dx = px[i] - lx;
            float dy = py[i] - ly;
            float dz = pz[i] - lz;
            // unfused (mul,mul,mul,add,add) rounding, exactly as the reference
            float d2 = __fadd_rn(__fadd_rn(__fmul_rn(dx, dx), __fmul_rn(dy, dy)),
                                 __fmul_rn(dz, dz));
            float dn = fminf(dist[i], d2);
            dist[i] = dn;
            // strict '>' keeps the first (smallest-p) max within this thread
            if (dn > bestd) { bestd = dn; bestp = (unsigned)(tid + i * 256); }
        }
        // packed key: non-negative float bits are order-preserving; inverted
        // index gives smallest-index tie-break (jnp.argmax first occurrence)
        unsigned long long best =
            ((unsigned long long)__float_as_uint(bestd) << 32) |
            (unsigned long long)(0xFFFFFFFFu - bestp);

        // wave32 butterfly reduction
#pragma unroll
        for (int s = 16; s > 0; s >>= 1) {
            unsigned long long o = __shfl_xor(best, s, 32);
            best = o > best ? o : best;
        }

        const int buf = m & 1;                       // double-buffered slots
        if ((tid & 31) == 0) skeys[buf][tid >> 5] = best;
        __syncthreads();                             // single barrier / iter

        unsigned long long k = skeys[buf][0];
#pragma unroll
        for (int w = 1; w < 8; ++w) {
            unsigned long long o = skeys[buf][w];
            k = o > k ? o : k;
        }
        last = (int)(0xFFFFFFFFu - (unsigned)(k & 0xFFFFFFFFull));
        if (tid == 0) idx_out[b * MSAMP + m] = b * NPER + last;
        // no second barrier: next iteration writes the other skeys buffer
    }
}

// ---------------------------------------------------------------------------
// Kernel 2: bandwidth-bound gather of x / pos / batch rows (64 MB stream)
// one block per sampled row; 128 lanes x b128 covers the 512-float row;
// all data is touched exactly once -> non-temporal load/store
// ---------------------------------------------------------------------------
__global__ __launch_bounds__(128)
void gather_kernel(const float* __restrict__ x, const float* __restrict__ pos,
                   const int* __restrict__ idx, float* __restrict__ out) {
    const int r = blockIdx.x;     // 0 .. 16383
    const int t = threadIdx.x;    // 0 .. 127
    const int src = idx[r];

    float* out_x     = out;                                    // [16384*512]
    float* out_pos   = out + (size_t)(NB * MSAMP) * DFEAT;     // [16384*3]
    float* out_batch = out_pos + (size_t)(NB * MSAMP) * 3;     // [16384]

    const v4f* xs = (const v4f*)x + (size_t)src * (DFEAT / 4);
    v4f*       xd = (v4f*)out_x   + (size_t)r   * (DFEAT / 4);
    v4f v = __builtin_nontemporal_load(&xs[t]);
    __builtin_nontemporal_store(v, &xd[t]);

    if (t < 3)  out_pos[r * 3 + t] = pos[src * 3 + t];
    if (t == 3) out_batch[r] = (float)(src >> 12);   // batch id = src / 4096
}

// ---------------------------------------------------------------------------
extern "C" void kernel_launch(void* const* d_in, const int* in_sizes, int n_in,
                              void* d_out, int out_size, void* d_ws, size_t ws_size,
                              hipStream_t stream) {
    const float* x   = (const float*)d_in[0];   // [65536, 512] f32
    const float* pos = (const float*)d_in[1];   // [65536, 3]   f32
    (void)in_sizes; (void)n_in; (void)ws_size; (void)out_size;

    int*   idx_ws = (int*)d_ws;                 // 16384 sampled global indices
    float* out    = (float*)d_out;

    fps_kernel<<<NB, 256, 0, stream>>>(pos, idx_ws);
    gather_kernel<<<NB * MSAMP, 128, 0, stream>>>(x, pos, idx_ws, out);
}